// ConvReshapeBefore_27934467293596
// MI455X (gfx1250) — compile-verified
//
#include <hip/hip_runtime.h>

// im2col patch extraction: x(32,64,64,64) f32 -> out(62*62*32, 3,3,64) f32
// out[((r*62+c)*32+b)*576 + kh*192 + kw*64 + ch] = x[(b*64 + r+kh)*4096 + (c+kw)*64 + ch]
// Pure data movement: ~283 MB stores + ~33.5 MB unique loads => HBM-bound (~13.6 us @ 23.3 TB/s).
// Plan: one WG per (b,kh,r). TDM DMA the 16KB input row (b, r+kh) into LDS once,
// then all waves emit the 62 overlapping 192-float chunks with B128 NT stores.

#define OR_    62
#define OC_    62
#define NB_    32
#define KH_    3
#define ROWLEN 4096        // W*C floats per (b,row)
#define NCHUNK4 2976       // 62 chunks * 48 float4 per chunk

typedef __attribute__((ext_vector_type(4))) float        f32x4;
typedef __attribute__((ext_vector_type(4))) unsigned int v4u;
typedef __attribute__((ext_vector_type(4))) int          v4i;
typedef __attribute__((ext_vector_type(8))) int          v8i;

#if __has_builtin(__builtin_amdgcn_tensor_load_to_lds) && __has_builtin(__builtin_amdgcn_s_wait_tensorcnt)
#define USE_TDM 1
#else
#define USE_TDM 0
#endif

__global__ __launch_bounds__(256) void ConvReshapeBefore_im2col_kernel(
    const float* __restrict__ x, float* __restrict__ out) {
  __shared__ float row_lds[ROWLEN];

  const int      tid = threadIdx.x;
  const unsigned wg  = blockIdx.x;
  const int r   = (int)(wg % OR_);
  const unsigned t2 = wg / OR_;
  const int kh  = (int)(t2 % KH_);
  const int b   = (int)(t2 / KH_);
  const int row = r + kh;                      // input H index, 0..63

  const float* row_src = x + ((size_t)(b * 64 + row) * ROWLEN);

#if USE_TDM
  // ---- Wave 0 issues one Tensor Data Mover DMA: 4096 f32 global -> LDS ----
  if (tid < 32) {
    const unsigned long long ga = (unsigned long long)(uintptr_t)row_src;
    // Flat LDS pointer: low 32 bits are the workgroup-relative LDS byte offset.
    const unsigned lds_off = (unsigned)(uintptr_t)row_lds;

    v4u g0;                                     // D# group 0 (128b)
    g0.x = 1u;                                  // count=1 (valid user descriptor)
    g0.y = lds_off;                             // lds_addr (bytes)
    g0.z = (unsigned)(ga & 0xFFFFFFFFull);      // global_addr[31:0]
    g0.w = (unsigned)((ga >> 32) & 0x01FFFFFFull) | (2u << 30); // addr[56:32] | type=2

    v8i g1;                                     // D# group 1 (256b)
    g1[0] = (int)(2u << 16);                    // workgroup_mask=0, data_size=2 (4 bytes)
    g1[1] = (int)((unsigned)(ROWLEN & 0xFFFF) << 16); // tensor_dim0[15:0]=4096
    g1[2] = (int)(1u << 16);                    // tensor_dim0 hi=0, tensor_dim1=1
    g1[3] = (int)((unsigned)(ROWLEN & 0xFFFF) << 16); // tile_dim0=4096
    g1[4] = 1;                                  // tile_dim1=1, tile_dim2=0
    g1[5] = ROWLEN;                             // tensor_dim0_stride=4096
    g1[6] = 0;                                  // stride hi / dim1_stride lo
    g1[7] = 0;                                  // dim1_stride hi

    v4i gz  = {0, 0, 0, 0};                     // groups 2/3 unused (<=2D tile)
    v8i gz8 = {0, 0, 0, 0, 0, 0, 0, 0};         // extra group (6-arg toolchain form)
    __builtin_amdgcn_tensor_load_to_lds(g0, g1, gz, gz, gz8, 0);
    __builtin_amdgcn_s_wait_tensorcnt(0);       // TENSORcnt drains before barrier
  }
  __syncthreads();
#else
  // Fallback: cooperative B128 copy global -> LDS.
  {
    const f32x4* s4 = (const f32x4*)row_src;
    f32x4*       d4 = (f32x4*)row_lds;
    for (int i = tid; i < ROWLEN / 4; i += 256) d4[i] = s4[i];
  }
  __syncthreads();
#endif

  // ---- Emit 62 chunks x 192 floats (48 float4 each) from LDS with NT B128 stores ----
  const f32x4* lds4 = (const f32x4*)row_lds;
  f32x4*       out4 = (f32x4*)out;
  // chunk c base (in float4 units): ((r*62+c)*32+b)*144 + kh*48
  const unsigned base0   = ((unsigned)(r * (OC_ * NB_)) + (unsigned)b) * 144u + (unsigned)kh * 48u;
  const unsigned cstride = (unsigned)NB_ * 144u;   // 4608 float4 per c-step

#pragma unroll
  for (int k = 0; k < 12; ++k) {
    int idx = tid + (k << 8);                   // 0..3071; 2976 valid
    if (idx < NCHUNK4) {
      int c = idx / 48;
      int j = idx - c * 48;                     // float4 within chunk (kw*16 + ch/4)
      f32x4 v = lds4[c * 16 + j];               // ds_load_b128, 16B aligned
      __builtin_nontemporal_store(v, &out4[base0 + (unsigned)c * cstride + (unsigned)j]);
    }
  }
}

extern "C" void kernel_launch(void* const* d_in, const int* in_sizes, int n_in,
                              void* d_out, int out_size, void* d_ws, size_t ws_size,
                              hipStream_t stream) {
  const float* x   = (const float*)d_in[0];
  float*       out = (float*)d_out;
  const int nblocks = NB_ * KH_ * OR_;          // 32*3*62 = 5952 workgroups
  ConvReshapeBefore_im2col_kernel<<<nblocks, 256, 0, stream>>>(x, out);
}